// SimpleGAT_28741921145426
// MI455X (gfx1250) — compile-verified
//
#include <hip/hip_runtime.h>
#include <hip/hip_bf16.h>
#include <math.h>

typedef float v2f __attribute__((ext_vector_type(2)));
typedef float v8f __attribute__((ext_vector_type(8)));

namespace {
constexpr int   N_  = 100000;
constexpr int   E_  = 1000000;
constexpr int   IN_ = 32;
constexpr int   H_  = 64;
constexpr int   ED_ = 16;
constexpr int   G_  = 512;
constexpr int   H3_ = 192;
constexpr int   EN_ = E_ + N_;
constexpr float SLOPE_ = 0.2f;
constexpr unsigned ENC_NEG_INF = 0x007fffffu; // enc(-inf)
}

__device__ __forceinline__ unsigned enc_f32(float f) {
  unsigned u = __float_as_uint(f);
  return (u & 0x80000000u) ? ~u : (u | 0x80000000u);
}
__device__ __forceinline__ float dec_f32(unsigned u) {
  return __uint_as_float((u & 0x80000000u) ? (u & 0x7fffffffu) : ~u);
}

// ---------------------------------------------------------------------------
// Generic fp32 WMMA GEMM: C[M x Ncols] = act(A[M x K] @ B[K x Ncols] + bias)
// One wave (32 lanes) per 16x16 output tile. grid = (M/16, Ncols/16).
// Uses V_WMMA_F32_16X16X4_F32 (fp32, exact accumulation).
// ---------------------------------------------------------------------------
template <int K, bool RELU, bool BIAS>
__global__ __launch_bounds__(32) void wmma_gemm_kernel(
    const float* __restrict__ A, const float* __restrict__ B,
    const float* __restrict__ bias, float* __restrict__ C, int Ncols) {
  const int lane = threadIdx.x;
  const int lm = lane & 15;   // M (for A) / N (for B/C) position
  const int lk = lane >> 4;   // K-half selector
  const int m0 = blockIdx.x * 16;
  const int n0 = blockIdx.y * 16;

  v8f acc = {};
  const float* __restrict__ Arow = A + (size_t)(m0 + lm) * K;
#pragma unroll
  for (int k0 = 0; k0 < K; k0 += 4) {
    v2f a, b;
    a.x = Arow[k0 + 2 * lk];
    a.y = Arow[k0 + 2 * lk + 1];
    const float* __restrict__ Bp = B + (size_t)(k0 + 2 * lk) * Ncols + n0 + lm;
    b.x = Bp[0];
    b.y = Bp[Ncols];
    acc = __builtin_amdgcn_wmma_f32_16x16x4_f32(false, a, false, b, (short)0,
                                                acc, false, false);
  }
  const float bv = BIAS ? bias[n0 + lm] : 0.0f;
#pragma unroll
  for (int r = 0; r < 8; ++r) {
    float v = acc[r] + bv;
    if (RELU) v = fmaxf(v, 0.0f);
    C[(size_t)(m0 + r + lk * 8) * Ncols + n0 + lm] = v;
  }
}

// ---------------------------------------------------------------------------
// Fused edge embedding: ea = relu(edge_attr @ w_edge + b_edge) for 16 edges
// (never stored globally). Produces: ae[i][edge] = ea . we_att[i] (3 scalars),
// esum[dst] += ea, cnt[dst] += 1.  One wave per 16 edges.
// ---------------------------------------------------------------------------
__global__ __launch_bounds__(32) void edge_embed_kernel(
    const float* __restrict__ EA, const float* __restrict__ Wedge,
    const float* __restrict__ bedge, const float* __restrict__ weatt,
    const int* __restrict__ dst, float* __restrict__ esum,
    float* __restrict__ cnt, float* __restrict__ ae) {
  __shared__ float tile[16 * H_];
  __shared__ float red[3][32];
  const int lane = threadIdx.x;
  const int lm = lane & 15, lk = lane >> 4;
  const int e0 = blockIdx.x * 16;

  v8f acc[4] = {};
  const float* __restrict__ Arow = EA + (size_t)(e0 + lm) * ED_;
#pragma unroll
  for (int k0 = 0; k0 < ED_; k0 += 4) {
    v2f a;
    a.x = Arow[k0 + 2 * lk];
    a.y = Arow[k0 + 2 * lk + 1];
#pragma unroll
    for (int t = 0; t < 4; ++t) {
      v2f b;
      const float* __restrict__ Bp = Wedge + (size_t)(k0 + 2 * lk) * H_ + t * 16 + lm;
      b.x = Bp[0];
      b.y = Bp[H_];
      acc[t] = __builtin_amdgcn_wmma_f32_16x16x4_f32(false, a, false, b,
                                                     (short)0, acc[t], false, false);
    }
  }
#pragma unroll
  for (int t = 0; t < 4; ++t)
#pragma unroll
    for (int r = 0; r < 8; ++r) {
      const int row = r + lk * 8;
      const int col = t * 16 + lm;
      tile[row * H_ + col] = fmaxf(acc[t][r] + bedge[col], 0.0f);
    }
  __syncthreads();

  // per-edge dot with we_att[0..2]; lane pair (lm, lm+16) splits 64 cols
  const int cbase = lk * 32;
  float p0 = 0.f, p1 = 0.f, p2 = 0.f;
#pragma unroll
  for (int c = 0; c < 32; ++c) {
    const float v = tile[lm * H_ + cbase + c];
    p0 += v * weatt[0 * H_ + cbase + c];
    p1 += v * weatt[1 * H_ + cbase + c];
    p2 += v * weatt[2 * H_ + cbase + c];
  }
  red[0][lane] = p0;
  red[1][lane] = p1;
  red[2][lane] = p2;
  __syncthreads();

  const int eIdx = e0 + lm;
  const int d = dst[eIdx];
  if (lk == 0) {
    ae[(size_t)0 * EN_ + eIdx] = red[0][lane] + red[0][lane + 16];
    ae[(size_t)1 * EN_ + eIdx] = red[1][lane] + red[1][lane + 16];
    ae[(size_t)2 * EN_ + eIdx] = red[2][lane] + red[2][lane + 16];
    atomicAdd(&cnt[d], 1.0f);
  }
#pragma unroll
  for (int c = 0; c < 32; ++c)
    atomicAdd(&esum[(size_t)d * H_ + cbase + c], tile[lm * H_ + cbase + c]);
}

// ---------------------------------------------------------------------------
// Fused h = xi @ conv_W  (no bias), plus s_score = h.att_s, d_score = h.att_d
// One wave per 16 nodes (full 16x64 tile).
// ---------------------------------------------------------------------------
__global__ __launch_bounds__(32) void h_scores_kernel(
    const float* __restrict__ xi, const float* __restrict__ Wl,
    const float* __restrict__ att_s, const float* __restrict__ att_d,
    float* __restrict__ h, float* __restrict__ sS, float* __restrict__ dS) {
  __shared__ float tile[16 * H_];
  __shared__ float red[2][32];
  const int lane = threadIdx.x;
  const int lm = lane & 15, lk = lane >> 4;
  const int m0 = blockIdx.x * 16;

  v8f acc[4] = {};
  const float* __restrict__ Arow = xi + (size_t)(m0 + lm) * H_;
#pragma unroll
  for (int k0 = 0; k0 < H_; k0 += 4) {
    v2f a;
    a.x = Arow[k0 + 2 * lk];
    a.y = Arow[k0 + 2 * lk + 1];
#pragma unroll
    for (int t = 0; t < 4; ++t) {
      v2f b;
      const float* __restrict__ Bp = Wl + (size_t)(k0 + 2 * lk) * H_ + t * 16 + lm;
      b.x = Bp[0];
      b.y = Bp[H_];
      acc[t] = __builtin_amdgcn_wmma_f32_16x16x4_f32(false, a, false, b,
                                                     (short)0, acc[t], false, false);
    }
  }
#pragma unroll
  for (int t = 0; t < 4; ++t)
#pragma unroll
    for (int r = 0; r < 8; ++r) {
      const int row = r + lk * 8;
      const int col = t * 16 + lm;
      const float v = acc[t][r];
      tile[row * H_ + col] = v;
      h[(size_t)(m0 + row) * H_ + col] = v;
    }
  __syncthreads();

  const int cbase = lk * 32;
  float p0 = 0.f, p1 = 0.f;
#pragma unroll
  for (int c = 0; c < 32; ++c) {
    const float v = tile[lm * H_ + cbase + c];
    p0 += v * att_s[cbase + c];
    p1 += v * att_d[cbase + c];
  }
  red[0][lane] = p0;
  red[1][lane] = p1;
  __syncthreads();
  if (lk == 0) {
    sS[m0 + lm] = red[0][lane] + red[0][lane + 16];
    dS[m0 + lm] = red[1][lane] + red[1][lane + 16];
  }
}

// ------------------------- small elementwise kernels ------------------------
__global__ void weatt_kernel(const float* __restrict__ We,
                             const float* __restrict__ atte,
                             float* __restrict__ weatt) {
  const int t = blockIdx.x * blockDim.x + threadIdx.x;
  if (t >= 3 * H_) return;
  const int i = t / H_, k = t % H_;
  float s = 0.f;
#pragma unroll
  for (int n = 0; n < H_; ++n)
    s += We[(size_t)i * H_ * H_ + (size_t)k * H_ + n] * atte[i * H_ + n];
  weatt[t] = s;
}

__global__ void pre_init_kernel(float* __restrict__ esum, float* __restrict__ cnt) {
  const int t = blockIdx.x * blockDim.x + threadIdx.x;
  if (t < N_ * H_) esum[t] = 0.0f;
  if (t < N_) cnt[t] = 0.0f;
}

__global__ void loop_feats_kernel(const float* __restrict__ esum,
                                  const float* __restrict__ cnt,
                                  const float* __restrict__ weatt,
                                  float* __restrict__ ae) {
  const int n = blockIdx.x * blockDim.x + threadIdx.x;
  if (n >= N_) return;
  const float c = cnt[n];
  float a0 = 0.f, a1 = 0.f, a2 = 0.f;
  if (c > 0.0f) {
    const float* __restrict__ row = esum + (size_t)n * H_;
#pragma unroll
    for (int k = 0; k < H_; ++k) {
      const float v = row[k];
      a0 += v * weatt[0 * H_ + k];
      a1 += v * weatt[1 * H_ + k];
      a2 += v * weatt[2 * H_ + k];
    }
    const float inv = 1.0f / c;
    a0 *= inv; a1 *= inv; a2 *= inv;
  }
  ae[(size_t)0 * EN_ + E_ + n] = a0;
  ae[(size_t)1 * EN_ + E_ + n] = a1;
  ae[(size_t)2 * EN_ + E_ + n] = a2;
}

__global__ void attn_init_kernel(float* __restrict__ agg, float* __restrict__ den,
                                 unsigned* __restrict__ mmax) {
  const int t = blockIdx.x * blockDim.x + threadIdx.x;
  if (t < N_ * H_) agg[t] = 0.0f;
  if (t < N_) { den[t] = 0.0f; mmax[t] = ENC_NEG_INF; }
}

__global__ void score1_kernel(const float* __restrict__ sS, const float* __restrict__ dS,
                              const float* __restrict__ aeL, const int* __restrict__ src,
                              const int* __restrict__ dst, float* __restrict__ aex,
                              unsigned* __restrict__ mmax) {
  const int idx = blockIdx.x * blockDim.x + threadIdx.x;
  if (idx >= EN_) return;
  int s, d;
  if (idx < E_) { s = src[idx]; d = dst[idx]; }
  else { s = idx - E_; d = s; }
  float a = sS[s] + dS[d] + aeL[idx];
  a = (a > 0.0f) ? a : SLOPE_ * a;   // leaky_relu
  aex[idx] = a;
  atomicMax(&mmax[d], enc_f32(a));
}

__global__ void score2_kernel(const int* __restrict__ dst, float* __restrict__ aex,
                              const unsigned* __restrict__ mmax,
                              float* __restrict__ den) {
  const int idx = blockIdx.x * blockDim.x + threadIdx.x;
  if (idx >= EN_) return;
  const int d = (idx < E_) ? dst[idx] : (idx - E_);
  float m = dec_f32(mmax[d]);
  if (!isfinite(m)) m = 0.0f;
  const float ex = expf(aex[idx] - m);
  aex[idx] = ex;
  atomicAdd(&den[d], ex);
}

__global__ void agg_kernel(const int* __restrict__ src, const int* __restrict__ dst,
                           const float* __restrict__ aex, const float* __restrict__ den,
                           const float* __restrict__ h, float* __restrict__ agg) {
  const int idx = blockIdx.x * blockDim.x + threadIdx.x;
  if (idx >= EN_) return;
  int s, d;
  if (idx < E_) { s = src[idx]; d = dst[idx]; }
  else { s = idx - E_; d = s; }
  const float alpha = aex[idx] / den[d];
  const float4* __restrict__ hp = (const float4*)(h + (size_t)s * H_);
  float* __restrict__ ap = agg + (size_t)d * H_;
#pragma unroll
  for (int c = 0; c < H_ / 4; ++c) {
    const float4 v = hp[c];
    atomicAdd(ap + 4 * c + 0, alpha * v.x);
    atomicAdd(ap + 4 * c + 1, alpha * v.y);
    atomicAdd(ap + 4 * c + 2, alpha * v.z);
    atomicAdd(ap + 4 * c + 3, alpha * v.w);
  }
}

__global__ void finalize_kernel(const float* __restrict__ agg,
                                const float* __restrict__ cb,
                                float* __restrict__ h) {
  const int t = blockIdx.x * blockDim.x + threadIdx.x;
  if (t >= N_ * H_) return;
  h[t] = fmaxf(agg[t] + cb[t % H_], 0.0f);
}

__global__ void gru_elem_kernel(const float* __restrict__ gi,
                                const float* __restrict__ gh,
                                float* __restrict__ xi) {
  const int t = blockIdx.x * blockDim.x + threadIdx.x;
  if (t >= N_ * H_) return;
  const int n = t / H_, c = t % H_;
  const size_t base = (size_t)n * H3_;
  const float ir = gi[base + c],        hr = gh[base + c];
  const float iz = gi[base + H_ + c],   hz = gh[base + H_ + c];
  const float in2 = gi[base + 2 * H_ + c], hn = gh[base + 2 * H_ + c];
  const float r = 1.0f / (1.0f + expf(-(ir + hr)));
  const float z = 1.0f / (1.0f + expf(-(iz + hz)));
  const float nn = tanhf(in2 + r * hn);
  const float v = (1.0f - z) * nn + z * xi[t];
  xi[t] = fmaxf(v, 0.0f);
}

__global__ void zero_kernel(float* __restrict__ p, int n) {
  const int t = blockIdx.x * blockDim.x + threadIdx.x;
  if (t < n) p[t] = 0.0f;
}

__global__ void pool_add_kernel(const float* __restrict__ xi,
                                const int* __restrict__ batch,
                                float* __restrict__ pooled) {
  const int t = blockIdx.x * blockDim.x + threadIdx.x;
  if (t >= N_ * H_) return;
  const int n = t / H_, c = t % H_;
  atomicAdd(&pooled[(size_t)batch[n] * H_ + c], xi[t]);
}

__global__ void final_out_kernel(const float* __restrict__ pooled,
                                 const float* __restrict__ w_out,
                                 const float* __restrict__ b_out,
                                 float* __restrict__ out) {
  const int g = blockIdx.x * blockDim.x + threadIdx.x;
  if (g >= G_) return;
  float acc = b_out[0];
#pragma unroll
  for (int k = 0; k < H_; ++k) acc += pooled[(size_t)g * H_ + k] * w_out[k];
  out[g] = acc;
}

// ---------------------------------------------------------------------------
extern "C" void kernel_launch(void* const* d_in, const int* in_sizes, int n_in,
                              void* d_out, int out_size, void* d_ws, size_t ws_size,
                              hipStream_t stream) {
  (void)in_sizes; (void)n_in; (void)out_size; (void)ws_size;
  const float* x         = (const float*)d_in[0];
  const float* edge_attr = (const float*)d_in[1];
  const float* w_node    = (const float*)d_in[2];
  const float* b_node    = (const float*)d_in[3];
  const float* w_edge    = (const float*)d_in[4];
  const float* b_edge    = (const float*)d_in[5];
  const float* conv_W    = (const float*)d_in[6];
  const float* conv_We   = (const float*)d_in[7];
  const float* att_s     = (const float*)d_in[8];
  const float* att_d     = (const float*)d_in[9];
  const float* att_e     = (const float*)d_in[10];
  const float* conv_b    = (const float*)d_in[11];
  const float* gru_Wih   = (const float*)d_in[12];
  const float* gru_Whh   = (const float*)d_in[13];
  const float* gru_bih   = (const float*)d_in[14];
  const float* gru_bhh   = (const float*)d_in[15];
  const float* w_out     = (const float*)d_in[16];
  const float* b_out     = (const float*)d_in[17];
  const int*   eidx      = (const int*)d_in[18];
  const int*   batch     = (const int*)d_in[19];
  const int* src = eidx;
  const int* dst = eidx + E_;
  float* out = (float*)d_out;

  float* W = (float*)d_ws;
  size_t o = 0;
  float* xi    = W + o; o += (size_t)N_ * H_;
  float* h     = W + o; o += (size_t)N_ * H_;
  float* esum  = W + o; o += (size_t)N_ * H_;
  float* agg   = W + o; o += (size_t)N_ * H_;
  float* gi    = W + o; o += (size_t)N_ * H3_;
  float* gh    = W + o; o += (size_t)N_ * H3_;
  float* ae    = W + o; o += (size_t)3 * EN_;
  float* aex   = W + o; o += EN_;
  float* sS    = W + o; o += N_;
  float* dS    = W + o; o += N_;
  float* cnt   = W + o; o += N_;
  float* den   = W + o; o += N_;
  unsigned* mmax = (unsigned*)(W + o); o += N_;
  float* pooled = W + o; o += (size_t)G_ * H_;
  float* weatt  = W + o; o += 3 * H_;

  const int NH = N_ * H_;
  const int blkNH = (NH + 255) / 256;
  const int blkEN = (EN_ + 255) / 256;

  // Stage 0: init + precompute we_att[i] = conv_We[i] @ att_e[i]
  pre_init_kernel<<<blkNH, 256, 0, stream>>>(esum, cnt);
  weatt_kernel<<<1, 3 * H_, 0, stream>>>(conv_We, att_e, weatt);

  // Stage 1: xi = relu(x @ w_node + b_node)   [WMMA fp32]
  wmma_gemm_kernel<IN_, true, true>
      <<<dim3(N_ / 16, H_ / 16), 32, 0, stream>>>(x, w_node, b_node, xi, H_);

  // Stage 2: fused edge embedding -> ae scalars, esum, cnt   [WMMA fp32]
  edge_embed_kernel<<<E_ / 16, 32, 0, stream>>>(edge_attr, w_edge, b_edge,
                                                weatt, dst, esum, cnt, ae);
  // Stage 3: self-loop attention scalars from mean in-edge features
  loop_feats_kernel<<<(N_ + 255) / 256, 256, 0, stream>>>(esum, cnt, weatt, ae);

  for (int layer = 0; layer < 3; ++layer) {
    const float* Wl   = conv_W  + (size_t)layer * H_ * H_;
    const float* asl  = att_s   + (size_t)layer * H_;
    const float* adl  = att_d   + (size_t)layer * H_;
    const float* cbl  = conv_b  + (size_t)layer * H_;
    const float* Wih  = gru_Wih + (size_t)layer * H_ * H3_;
    const float* Whh  = gru_Whh + (size_t)layer * H_ * H3_;
    const float* bih  = gru_bih + (size_t)layer * H3_;
    const float* bhh  = gru_bhh + (size_t)layer * H3_;
    const float* aeL  = ae + (size_t)layer * EN_;

    // h = xi @ W, plus per-node attention scores   [WMMA fp32]
    h_scores_kernel<<<N_ / 16, 32, 0, stream>>>(xi, Wl, asl, adl, h, sS, dS);

    // segment softmax over destinations (incl. self loops)
    attn_init_kernel<<<blkNH, 256, 0, stream>>>(agg, den, mmax);
    score1_kernel<<<blkEN, 256, 0, stream>>>(sS, dS, aeL, src, dst, aex, mmax);
    score2_kernel<<<blkEN, 256, 0, stream>>>(dst, aex, mmax, den);
    agg_kernel<<<blkEN, 256, 0, stream>>>(src, dst, aex, den, h, agg);
    finalize_kernel<<<blkNH, 256, 0, stream>>>(agg, cbl, h);

    // GRU gates   [WMMA fp32]
    wmma_gemm_kernel<H_, false, true>
        <<<dim3(N_ / 16, H3_ / 16), 32, 0, stream>>>(h, Wih, bih, gi, H3_);
    wmma_gemm_kernel<H_, false, true>
        <<<dim3(N_ / 16, H3_ / 16), 32, 0, stream>>>(xi, Whh, bhh, gh, H3_);
    gru_elem_kernel<<<blkNH, 256, 0, stream>>>(gi, gh, xi);
  }

  // global_add_pool + output projection
  zero_kernel<<<(G_ * H_ + 255) / 256, 256, 0, stream>>>(pooled, G_ * H_);
  pool_add_kernel<<<blkNH, 256, 0, stream>>>(xi, batch, pooled);
  final_out_kernel<<<(G_ + 255) / 256, 256, 0, stream>>>(pooled, w_out, b_out, out);
}